// ContrastiveLoss_25769803776292
// MI455X (gfx1250) — compile-verified
//
#include <hip/hip_runtime.h>
#include <hip/hip_bf16.h>
#include <math.h>

typedef __attribute__((ext_vector_type(2))) float v2f;
typedef __attribute__((ext_vector_type(8))) float v8f;

// ---------------------------------------------------------------------------
// Kernel 1: build pos_idx / neg_idx (sorted, like np.nonzero) + header.
// Single block, 1024 threads, Hillis-Steele scan of per-thread pos counts.
// y in {0,1}: neg rank of element i == i - (#pos before i).
// ---------------------------------------------------------------------------
__global__ void __launch_bounds__(1024)
scan_build_idx(const int* __restrict__ y, int HW,
               int* __restrict__ hdr, float* __restrict__ sbuf,
               int* __restrict__ posIdx, int* __restrict__ negIdx) {
    __shared__ int lds[1024];
    const int t = threadIdx.x;
    const int nthr = blockDim.x;
    const int chunk = HW / nthr;
    const int start = t * chunk;

    int cnt = 0;
    for (int i = 0; i < chunk; ++i) cnt += (y[start + i] == 1);

    lds[t] = cnt;
    __syncthreads();
    for (int off = 1; off < nthr; off <<= 1) {
        int v = (t >= off) ? lds[t - off] : 0;
        __syncthreads();
        lds[t] += v;
        __syncthreads();
    }
    const int incl  = lds[t];
    const int total = lds[nthr - 1];

    int rank = incl - cnt;                       // exclusive pos prefix
    for (int i = 0; i < chunk; ++i) {
        const int gi = start + i;
        if (y[gi] == 1) { posIdx[rank] = gi; ++rank; }
        else            { negIdx[gi - rank] = gi; }
    }

    if (t == 0) {
        const int Npos = total, Nneg = HW - total;
        const int P = Npos >> 1, Ng = Nneg >> 1;
        const int M = P < Ng ? P : Ng;
        hdr[0] = Npos; hdr[1] = Nneg; hdr[2] = P; hdr[3] = Ng; hdr[4] = M;
        sbuf[0] = 0.0f; sbuf[1] = 0.0f; sbuf[2] = 0.0f; sbuf[3] = 0.0f;
    }
}

// ---------------------------------------------------------------------------
// Kernel 2: batched column-pair dot products via V_WMMA_F32_16X16X4_F32.
// One wave = 16 pairs of one type (0:pos_sim 1:neg_sim 2:pn1 3:pn2).
// A fragment (16x4 f32): lane l -> row m=l&15, K-half = l>>4 (K=2h, 2h+1).
// B fragment (4x16 f32): lane l -> col n=l&15, same K-half (symmetric layout).
// 64 chained WMMAs accumulate the 16x16 cross-dot matrix; we use its diagonal.
//
// Addressing: uniform 64-bit base (advanced by a scalar add each K-chunk) +
// per-lane 32-bit byte offset that is LOOP-INVARIANT -> SADDR-form
// global_load_b32 with the 2nd channel folded into the 24-bit imm offset.
// ---------------------------------------------------------------------------
__global__ void __launch_bounds__(256)
pair_dot_kernel(const float* __restrict__ x, int HW, int C, int B,
                const int* __restrict__ hdr, float* __restrict__ sbuf,
                const int* __restrict__ posIdx, const int* __restrict__ negIdx,
                float* __restrict__ pn1, float* __restrict__ pn2) {
    const int lane = threadIdx.x & 31;
    const int w    = blockIdx.x * (blockDim.x >> 5) + (threadIdx.x >> 5);

    const int P  = hdr[2];
    const int Ng = hdr[3];
    const int M  = hdr[4];
    const int nG0 = (P  + 15) >> 4;
    const int nG1 = (Ng + 15) >> 4;
    const int nG2 = (M  + 15) >> 4;
    const int total = nG0 + nG1 + 2 * nG2;
    if (w >= total) return;                       // wave-uniform exit

    int type, g;
    if      (w < nG0)             { type = 0; g = w; }
    else if (w < nG0 + nG1)       { type = 1; g = w - nG0; }
    else if (w < nG0 + nG1 + nG2) { type = 2; g = w - nG0 - nG1; }
    else                          { type = 3; g = w - nG0 - nG1 - nG2; }

    const int m = lane & 15;                      // pair slot within group
    const int limit = (type == 0) ? P : (type == 1) ? Ng : M;
    const int i  = g * 16 + m;
    const bool valid = (i < limit);
    const int ic = valid ? i : (limit - 1);       // clamp: loads stay in-bounds

    int colL, colR;
    if      (type == 0) { colL = posIdx[ic];     colR = posIdx[P  + ic]; }
    else if (type == 1) { colL = negIdx[ic];     colR = negIdx[Ng + ic]; }
    else if (type == 2) { colL = posIdx[ic];     colR = negIdx[ic];      }
    else                { colL = posIdx[M + ic]; colR = negIdx[M + ic];  }

    const int khalf = lane >> 4;
    const unsigned planeB = (unsigned)HW * 4u;    // bytes per channel plane
    // Loop-invariant per-lane byte offsets (fit easily in 32 bits):
    const unsigned offL = (unsigned)(2 * khalf) * planeB + (unsigned)colL * 4u;
    const unsigned offR = (unsigned)(2 * khalf) * planeB + (unsigned)colR * 4u;
    const int nChunks = C >> 2;                   // K-chunks of 4 channels

    for (int bt = 0; bt < B; ++bt) {
        // Uniform base pointer, advanced by a scalar add per K-chunk.
        const char* pb = (const char*)(x + (size_t)bt * (size_t)C * (size_t)HW);

        v8f acc = {};
        #pragma unroll 4
        for (int kk = 0; kk < nChunks; ++kk) {
            v2f a, b;
            a.x = *(const float*)(pb + offL);             // K = 2h
            a.y = *(const float*)(pb + offL + planeB);    // K = 2h+1 (imm off)
            b.x = *(const float*)(pb + offR);
            b.y = *(const float*)(pb + offR + planeB);
            acc = __builtin_amdgcn_wmma_f32_16x16x4_f32(
                false, a, false, b, (short)0, acc, false, false);
            pb += (size_t)4 * planeB;                     // uniform advance
        }

        // Diagonal extraction from 16x16 f32 D layout:
        //   m<8 : VGPR m,   lane m        (own lane)
        //   m>=8: VGPR m-8, lane m+16     (fetch via shfl from upper half)
        const int idx = (lane < 16) ? (lane & 7) : ((lane - 24) & 7);
        float vsel =
            idx == 0 ? acc[0] : idx == 1 ? acc[1] : idx == 2 ? acc[2] :
            idx == 3 ? acc[3] : idx == 4 ? acc[4] : idx == 5 ? acc[5] :
            idx == 6 ? acc[6] : acc[7];
        const float shifted = __shfl(vsel, (lane + 16) & 31, 32);
        const float val = (lane < 8) ? vsel : shifted;

        const bool act = (lane < 16) && valid;
        if (type <= 1) {
            float vz = act ? val : 0.0f;
            for (int off = 16; off > 0; off >>= 1)
                vz += __shfl_down(vz, off, 32);
            if (lane == 0) unsafeAtomicAdd(&sbuf[type], vz);
        } else if (act) {
            float* dst = (type == 2) ? pn1 : pn2;
            dst[(size_t)bt * (size_t)M + (size_t)i] = val;
        }
    }
}

// ---------------------------------------------------------------------------
// Kernel 3: logsumexp over pn1/pn2 + combine means -> scalar loss.
// ---------------------------------------------------------------------------
__global__ void __launch_bounds__(1024)
finalize_kernel(const int* __restrict__ hdr, const float* __restrict__ sbuf,
                const float* __restrict__ pn1, const float* __restrict__ pn2,
                int B, float* __restrict__ out) {
    __shared__ float red[1024];
    const int t = threadIdx.x, nthr = blockDim.x;
    const int P = hdr[2], Ng = hdr[3], M = hdr[4];
    const int N = B * M;

    float lse[2];
    const float* bufs[2] = { pn1, pn2 };
    for (int k = 0; k < 2; ++k) {
        const float* pn = bufs[k];
        float mx = -INFINITY;
        for (int i = t; i < N; i += nthr) mx = fmaxf(mx, pn[i]);
        red[t] = mx; __syncthreads();
        for (int off = nthr >> 1; off > 0; off >>= 1) {
            if (t < off) red[t] = fmaxf(red[t], red[t + off]);
            __syncthreads();
        }
        mx = red[0]; __syncthreads();

        float s = 0.0f;
        for (int i = t; i < N; i += nthr) s += expf(pn[i] - mx);
        red[t] = s; __syncthreads();
        for (int off = nthr >> 1; off > 0; off >>= 1) {
            if (t < off) red[t] += red[t + off];
            __syncthreads();
        }
        const float sum = red[0]; __syncthreads();
        lse[k] = mx + logf(sum);
    }

    if (t == 0) {
        const float Spos = sbuf[0], Sneg = sbuf[1];
        const float loss = -Spos / (float)(B * P)
                         - Sneg / (float)(B * Ng)
                         + lse[0] + lse[1]
                         - 2.0f * logf((float)N);
        out[0] = loss;
    }
}

// ---------------------------------------------------------------------------
extern "C" void kernel_launch(void* const* d_in, const int* in_sizes, int n_in,
                              void* d_out, int out_size, void* d_ws, size_t ws_size,
                              hipStream_t stream) {
    const float* x = (const float*)d_in[0];
    const int*   y = (const int*)d_in[1];

    const int HW = in_sizes[1];                 // 65536
    const int B  = 4;
    const int C  = (int)((long long)in_sizes[0] / HW / B);   // 256

    // Workspace layout (bytes):
    //   [0,64)    : int header {Npos,Nneg,P,Ng,M,...}
    //   [64,128)  : float accumulators {S_pos,S_neg,_,_}
    //   [128, +4HW)          : pos_idx (worst case HW ints)
    //   [+4HW, +8HW)         : neg_idx
    //   [+8HW, +12HW)        : pn1 (B*M <= HW floats)
    //   [+12HW, +16HW)       : pn2
    char* ws = (char*)d_ws;
    int*   hdr    = (int*)ws;
    float* sbuf   = (float*)(ws + 64);
    int*   posIdx = (int*)(ws + 128);
    int*   negIdx = (int*)(ws + 128 + (size_t)HW * 4);
    float* pn1    = (float*)(ws + 128 + (size_t)HW * 8);
    float* pn2    = (float*)(ws + 128 + (size_t)HW * 12);

    scan_build_idx<<<1, 1024, 0, stream>>>(y, HW, hdr, sbuf, posIdx, negIdx);

    // Worst-case wave count: ceil((P+Ng)/16) + 2*ceil(M/16), P+Ng<=HW/2, M<=HW/4.
    const int maxWaves = ((HW / 2 + 15) / 16) + 2 * ((HW / 4 + 15) / 16) + 8;
    const int blocks   = (maxWaves + 7) / 8;     // 8 waves (256 thr) per block
    pair_dot_kernel<<<blocks, 256, 0, stream>>>(x, HW, C, B, hdr, sbuf,
                                                posIdx, negIdx, pn1, pn2);

    finalize_kernel<<<1, 1024, 0, stream>>>(hdr, sbuf, pn1, pn2, B,
                                            (float*)d_out);
}